// MultiHeadAttentionLayer_21388937134336
// MI455X (gfx1250) — compile-verified
//
#include <hip/hip_runtime.h>

// Problem constants from the reference: B=4, N=256, D=128, H=8, K=16
#define Bsz   4
#define Nn    256
#define Dd    128
#define Hh    8
#define Kk    16
#define HK    128          // H*K
#define Mrows 1024         // B*N

typedef __attribute__((ext_vector_type(2))) float v2f;
typedef __attribute__((ext_vector_type(8))) float v8f;

// ---------------------------------------------------------------------------
// Kernel 1: QKV = h @ [Wq | Wk | Wv] using V_WMMA_F32_16X16X4_F32 (fp32 WMMA).
// Grid: x = 24 (mat*8 + colTile, pure shift/mask decode), y = 8 (rowTile/8).
// Wave w of a block computes rowTile = blockIdx.y*8 + w; all 8 waves share
// the same W-column tile (B operand) for WGP$ reuse. 24*8 blocks * 8 waves
// = 1536 tiles exactly -> no bounds checks, EXEC all-ones (WMMA requirement).
// Kh output is pre-scaled by K^-0.5 = 0.25.
// ---------------------------------------------------------------------------
__global__ __launch_bounds__(256) void qkv_wmma_kernel(
    const float* __restrict__ h,
    const float* __restrict__ Wq,
    const float* __restrict__ Wk,
    const float* __restrict__ Wv,
    float* __restrict__ qkv /* [3][1024][128] in d_ws */)
{
  const int wave = threadIdx.x >> 5;   // 0..7
  const int lane = threadIdx.x & 31;
  const int half = lane >> 4;          // 0 or 1 (selects K 0/1 vs 2/3)
  const int lr   = lane & 15;          // row (A) / col (B,C) within tile

  const int mat     = blockIdx.x >> 3;        // 0=Q, 1=K, 2=V
  const int colTile = blockIdx.x & 7;
  const int rowTile = (blockIdx.y << 3) + wave;

  const float* W = (mat == 0) ? Wq : (mat == 1) ? Wk : Wv;
  const int row0 = rowTile << 4;
  const int col0 = colTile << 4;

  v8f c = {};
  for (int kk = 0; kk < Dd; kk += 4) {
    // A tile 16x4 (MxK): lane lr = row; VGPR0/1 hold K = 2*half / 2*half+1
    v2f a, b;
    const float* ap = h + (size_t)(row0 + lr) * Dd + (kk + 2 * half);
    a.x = ap[0];
    a.y = ap[1];
    // B tile 4x16 (KxN): lane lr = col; VGPR0/1 hold K = 2*half / 2*half+1
    const float* bp = W + (size_t)(kk + 2 * half) * HK + (col0 + lr);
    b.x = bp[0];
    b.y = bp[HK];
    // 8 args: (neg_a, A, neg_b, B, c_mod, C, reuse_a, reuse_b)
    c = __builtin_amdgcn_wmma_f32_16x16x4_f32(
        false, a, false, b, (short)0, c, false, false);
  }

  const float scale = (mat == 1) ? 0.25f : 1.0f;  // K^-0.5 folded into Kh
  float* outbase = qkv + (size_t)mat * Mrows * HK;
  // C/D layout: VGPR r, lanes 0-15 -> M=r, N=lane; lanes 16-31 -> M=r+8.
#pragma unroll
  for (int r = 0; r < 8; ++r) {
    const int row = row0 + r + 8 * half;
    outbase[(size_t)row * HK + (col0 + lr)] = c[r] * scale;
  }
}

// ---------------------------------------------------------------------------
// Kernel 2: the bandwidth-bound edge-attention stream (268 MB of e_att/e_val
// at 23.3 TB/s ~= 11.5 us floor). grid = B*N blocks (one per (b,i)),
// 128 threads (t = h*16 + k, 4 waves).
// - e_att/e_val: read-once -> non-temporal loads (TH=NT) so the stream does
//   not evict the L2-hot Q/Kh/V working set (re-read 256x per pass).
// - Kh/V: regular loads, L2 resident (1.5 MB << 192 MB L2).
// - Q.Kh dot over K=16 via shfl_xor tree inside each 16-lane half-wave.
// ---------------------------------------------------------------------------
__global__ __launch_bounds__(128) void edge_attn_kernel(
    const float* __restrict__ qkv,
    const float* __restrict__ e_att,
    const float* __restrict__ e_val,
    const float* __restrict__ mask,
    float* __restrict__ out)
{
  const int bi = blockIdx.x;        // b*N + i
  const int b  = bi >> 8;           // / N
  const int t  = threadIdx.x;       // h*16 + k

  const float* Q  = qkv;
  const float* Kh = qkv + (size_t)Mrows * HK;
  const float* V  = qkv + (size_t)2 * Mrows * HK;

  const float q  = Q[(size_t)bi * HK + t];
  const float mi = mask[bi];
  const int   bn = b * Nn;
  const size_t ebase = (size_t)bi * Nn * HK;

  float denom = 0.0f;
  float acc   = 0.0f;

#pragma unroll 4
  for (int j = 0; j < Nn; ++j) {
    const size_t rj   = (size_t)(bn + j) * HK + t;
    const size_t eidx = ebase + (size_t)j * HK + t;

    // Prefetch the HBM stream 8 iterations (8 KB/block) ahead; speculative
    // prefetch is dropped silently if it runs past the allocation.
    __builtin_prefetch(e_att + eidx + 8 * HK, 0, 0);
    __builtin_prefetch(e_val + eidx + 8 * HK, 0, 0);

    const float kh = Kh[rj];
    const float v  = V[rj];
    // Read-once stream: non-temporal so it doesn't pollute L2.
    const float ea = __builtin_nontemporal_load(e_att + eidx);
    const float ev = __builtin_nontemporal_load(e_val + eidx);

    // dot(Q[b,i,h,:], Kh[b,j,h,:]) over K=16 lanes of this half-wave
    float p = q * kh;
    p += __shfl_xor(p, 1);
    p += __shfl_xor(p, 2);
    p += __shfl_xor(p, 4);
    p += __shfl_xor(p, 8);

    float x = p + ea;
    x = fminf(5.0f, fmaxf(-5.0f, x));
    const float ex = __expf(x);

    const float m  = mi * mask[bn + j];
    const float sm = ex * m;       // scores * mask  (denominator term)
    denom += sm;
    acc   += sm * m * (v + ev);    // masked twice, matching the reference
  }

  out[(size_t)bi * HK + t] = acc / fmaxf(denom, 1e-6f);
}

// ---------------------------------------------------------------------------
// Launcher. Inputs (setup_inputs order): h, e_att, e_val, mask, Wq, Wk, Wv.
// d_ws holds Q|Kh|V = 3 * 1024 * 128 floats = 1.5 MB.
// ---------------------------------------------------------------------------
extern "C" void kernel_launch(void* const* d_in, const int* in_sizes, int n_in,
                              void* d_out, int out_size, void* d_ws, size_t ws_size,
                              hipStream_t stream) {
  const float* h     = (const float*)d_in[0];
  const float* e_att = (const float*)d_in[1];
  const float* e_val = (const float*)d_in[2];
  const float* mask  = (const float*)d_in[3];
  const float* Wq    = (const float*)d_in[4];
  const float* Wk    = (const float*)d_in[5];
  const float* Wv    = (const float*)d_in[6];
  float* out = (float*)d_out;
  float* qkv = (float*)d_ws;

  qkv_wmma_kernel<<<dim3(24, 8), 256, 0, stream>>>(h, Wq, Wk, Wv, qkv);
  edge_attn_kernel<<<Bsz * Nn, 128, 0, stream>>>(qkv, e_att, e_val, mask, out);
}